// ScaledDotProductAttention_73194832659127
// MI455X (gfx1250) — compile-verified
//
#include <hip/hip_runtime.h>
#include <hip/hip_bf16.h>
#include <stdint.h>

// ---------------------------------------------------------------------------
// Fused single-pass attention pooling for MI455X (gfx1250).
//   query = H[b, -1, :];  scores = H·q / 32;  w = softmax(scores);  out = wT·H
// H is 1 GiB fp32 -> purely HBM-bandwidth-bound (23.3 TB/s => ~46us floor for
// one pass). We stream H exactly once: async-copy 16x1024 fp32 tiles into LDS
// (double buffered, ASYNCcnt, non-temporal since data is single-use), compute
// the 16 scores with chained v_wmma_f32_16x16x4_f32 (query replicated across
// M; each wave owns a 128-wide K slice), then do an online-softmax weighted
// accumulate from the same LDS tile. Split-K over the sequence (16
// chunks/batch) for occupancy; a tiny second kernel merges the per-chunk
// (m, l, acc) partials.
// ---------------------------------------------------------------------------

#define HS 1024
#define SEQ 4096
#define NB 64
#define NCHUNK 16
#define CHUNK_ROWS (SEQ / NCHUNK)        // 256 rows per block
#define TILE_ROWS 16
#define NTILES (CHUNK_ROWS / TILE_ROWS)  // 16 tiles per block
#define WSS 1032                         // floats per (b,chunk) record
#define INV_SCALE 0.03125f               // 1/sqrt(1024)

typedef float v2f __attribute__((ext_vector_type(2)));
typedef float v8f __attribute__((ext_vector_type(8)));

__device__ __forceinline__ void wait_async0() {
#if __has_builtin(__builtin_amdgcn_s_wait_asynccnt)
  __builtin_amdgcn_s_wait_asynccnt(0);
#else
  asm volatile("s_wait_asynccnt 0" ::: "memory");
#endif
}

// 16 per-lane 16B async global->LDS copies sharing one base address pair.
// The 24-bit IOFFSET immediate is applied by HW to BOTH the LDS and global
// address (CDNA5 ISA), so no per-copy address math is needed.
// TH_LOAD_NT: the 1 GiB stream is single-use -> keep it out of L2.
template <int P>
struct AsyncTile {
  static __device__ __forceinline__ void run(uint32_t lds_off, const void* gaddr) {
    asm volatile("global_load_async_to_lds_b128 %0, %1, off offset:%2 th:TH_LOAD_NT"
                 :: "v"(lds_off), "v"(gaddr), "i"(P * 4096)
                 : "memory");
    AsyncTile<P + 1>::run(lds_off, gaddr);
  }
};
template <>
struct AsyncTile<16> {
  static __device__ __forceinline__ void run(uint32_t, const void*) {}
};

__global__ __launch_bounds__(256) void attn_partial_kernel(
    const float* __restrict__ H, float* __restrict__ ws) {
  extern __shared__ float smem[];
  float* tile0   = smem;                       // 16*1024 floats (64 KB)
  float* tile1   = smem + TILE_ROWS * HS;      // 16*1024 floats (64 KB)
  float* sc_part = smem + 2 * TILE_ROWS * HS;  // [8 waves][16 rows]

  const int tid  = threadIdx.x;
  const int w    = tid >> 5;   // wave id 0..7 (wave32)
  const int lane = tid & 31;
  const int b    = blockIdx.x / NCHUNK;
  const int c    = blockIdx.x % NCHUNK;

  const float* Hb     = H + (size_t)b * SEQ * HS;
  const float* gchunk = Hb + (size_t)c * CHUNK_ROWS * HS;
  const float* q      = Hb + (size_t)(SEQ - 1) * HS;

  // A-matrix fragments: A[m][k] = q[kbase+k] for all 16 rows m (replicated).
  // ISA 32-bit 16x4 A layout: lanes 0-15 hold K=0 (V0), K=1 (V1);
  // lanes 16-31 hold K=2 (V0), K=3 (V1).
  const int khalf = (lane < 16) ? 0 : 2;
  v2f afrag[32];
#pragma unroll
  for (int j = 0; j < 32; ++j) {
    const int kb = w * 128 + j * 4 + khalf;
    v2f a;
    a.x = q[kb];
    a.y = q[kb + 1];
    afrag[j] = a;
  }

  const uint32_t lds0 = (uint32_t)(uintptr_t)tile0 + (uint32_t)tid * 16u;
  const uint32_t lds1 = (uint32_t)(uintptr_t)tile1 + (uint32_t)tid * 16u;

  // Prologue: async-copy tile 0 (contiguous 64 KB) into buffer 0.
  AsyncTile<0>::run(lds0, (const char*)gchunk + tid * 16);

  float m = -3.0e38f;   // running max
  float l = 0.0f;       // running exp-sum
  float acc0 = 0.f, acc1 = 0.f, acc2 = 0.f, acc3 = 0.f;  // this thread's 4 cols
  const int col = tid * 4;

  for (int t = 0; t < NTILES; ++t) {
    const float* cbuf = (t & 1) ? tile1 : tile0;

    wait_async0();
    __syncthreads();  // barrier A: tile t in LDS; prev readers of other buf done

    if (t + 1 < NTILES) {
      const char* g = (const char*)(gchunk + (size_t)(t + 1) * TILE_ROWS * HS) + tid * 16;
      AsyncTile<0>::run(((t + 1) & 1) ? lds1 : lds0, g);
    }

    // --- score phase: D[m][n] = sum_k q[k]*H[row n][k] over this wave's
    //     128-wide K slice, via 32 chained v_wmma_f32_16x16x4_f32.
    // B layout (4x16 row-striped): V0 = K=0 (lanes 0-15) / K=2 (lanes 16-31),
    // V1 = K=1 / K=3  ->  lane loads H[lane%16][kb+khalf .. +1] as one b64.
    const float* bp = cbuf + (lane & 15) * HS + khalf;
    v8f cf = {};
#pragma unroll
    for (int j = 0; j < 32; ++j) {
      const v2f bf = *(const v2f*)(bp + w * 128 + j * 4);
      cf = __builtin_amdgcn_wmma_f32_16x16x4_f32(
          /*neg_a=*/false, afrag[j], /*neg_b=*/false, bf,
          /*c_mod=*/(short)0, cf, /*reuse_a=*/false, /*reuse_b=*/false);
    }
    // Every lane now holds this wave's partial score of row (lane%16).
    if (lane < 16) sc_part[w * 16 + lane] = cf[0];
    __syncthreads();  // barrier B: wave partials visible

    // --- all 256 threads: reduce 8 K-slice partials, online softmax,
    //     accumulate this thread's 4 private columns from the LDS tile.
    float s[TILE_ROWS];
    float tmax = -3.0e38f;
#pragma unroll
    for (int r = 0; r < TILE_ROWS; ++r) {
      float sr = 0.f;
#pragma unroll
      for (int ww = 0; ww < 8; ++ww) sr += sc_part[ww * 16 + r];
      sr *= INV_SCALE;
      s[r] = sr;
      tmax = fmaxf(tmax, sr);
    }
    const float mnew = fmaxf(m, tmax);
    const float corr = __expf(m - mnew);
    m = mnew;
    l *= corr;
    acc0 *= corr; acc1 *= corr; acc2 *= corr; acc3 *= corr;
#pragma unroll
    for (int r = 0; r < TILE_ROWS; ++r) {
      const float p = __expf(s[r] - mnew);
      l += p;
      const float4 h4 = *(const float4*)(cbuf + r * HS + col);
      acc0 = fmaf(p, h4.x, acc0);
      acc1 = fmaf(p, h4.y, acc1);
      acc2 = fmaf(p, h4.z, acc2);
      acc3 = fmaf(p, h4.w, acc3);
    }
  }

  // Emit per-chunk partial: [0]=m, [1]=l, [8..1031]=unnormalized accumulator.
  float* rec = ws + (size_t)(b * NCHUNK + c) * WSS;
  if (tid == 0) { rec[0] = m; rec[1] = l; }
  *(float4*)(rec + 8 + col) = make_float4(acc0, acc1, acc2, acc3);
}

__global__ __launch_bounds__(256) void attn_combine_kernel(
    const float* __restrict__ ws, float* __restrict__ out) {
  const int b   = blockIdx.x;
  const int tid = threadIdx.x;
  const float* recs = ws + (size_t)b * NCHUNK * WSS;

  float m = -3.0e38f;
#pragma unroll
  for (int c = 0; c < NCHUNK; ++c) m = fmaxf(m, recs[(size_t)c * WSS]);

  float wf[NCHUNK];
  float denom = 0.f;
#pragma unroll
  for (int c = 0; c < NCHUNK; ++c) {
    const float mc = recs[(size_t)c * WSS];
    const float lc = recs[(size_t)c * WSS + 1];
    wf[c] = __expf(mc - m);
    denom = fmaf(wf[c], lc, denom);
  }
  const float inv = 1.0f / denom;

  const int col = tid * 4;
  float a0 = 0.f, a1 = 0.f, a2 = 0.f, a3 = 0.f;
#pragma unroll
  for (int c = 0; c < NCHUNK; ++c) {
    const float4 h4 = *(const float4*)(recs + (size_t)c * WSS + 8 + col);
    a0 = fmaf(wf[c], h4.x, a0);
    a1 = fmaf(wf[c], h4.y, a1);
    a2 = fmaf(wf[c], h4.z, a2);
    a3 = fmaf(wf[c], h4.w, a3);
  }
  *(float4*)(out + (size_t)b * HS + col) =
      make_float4(a0 * inv, a1 * inv, a2 * inv, a3 * inv);
}

extern "C" void kernel_launch(void* const* d_in, const int* in_sizes, int n_in,
                              void* d_out, int out_size, void* d_ws, size_t ws_size,
                              hipStream_t stream) {
  const float* H = (const float*)d_in[0];
  float* out = (float*)d_out;
  float* ws  = (float*)d_ws;  // needs 64*16*1032*4 B ~= 4.2 MB

  // Dynamic LDS: 2 double-buffered 16x1024 fp32 tiles + 8x16 score partials.
  const size_t lds_bytes = (size_t)(2 * TILE_ROWS * HS + 8 * 16) * sizeof(float);

  attn_partial_kernel<<<dim3(NB * NCHUNK), dim3(256), lds_bytes, stream>>>(H, ws);
  attn_combine_kernel<<<dim3(NB), dim3(256), 0, stream>>>(ws, out);
}